// PinUtilization_16561393894025
// MI455X (gfx1250) — compile-verified
//
#include <hip/hip_runtime.h>

// ---- layout constants (match reference) ----
#define NBX 512
#define NBY 512
// bin size = 1.0, XL = YL = 0.0
#define PIN_STRETCH 1.4142135f
// 1/(BSX*BSY*UNIT_PIN_CAPACITY) = 10.0
#define OUT_SCALE 10.0f
// stretched extent < 2.0 -> at most 3 bins per axis carry positive overlap
// (reference's K=4 4th bin always has overlap <= 0 -> contributes exactly 0)
#define KCOV 3

__global__ __launch_bounds__(256) void PinUtil_scatter_kernel(
    const float2* __restrict__ sizes,   // [N] (w,h)
    const float2* __restrict__ pos,     // [N] (x,y)
    const float*  __restrict__ pinw,    // [N]
    float* __restrict__ grid,           // [NBX*NBY]
    int n)
{
    int i = blockIdx.x * blockDim.x + threadIdx.x;
    if (i >= n) return;

    float2 s = sizes[i];   // global_load_b64, coalesced
    float2 p = pos[i];     // global_load_b64, coalesced
    float  w = pinw[i];    // global_load_b32, coalesced

    float sw = fmaxf(s.x, PIN_STRETCH);
    float sh = fmaxf(s.y, PIN_STRETCH);
    float dens = (w * OUT_SCALE) / (sw * sh);

    float xlo = p.x - 0.5f * sw, xhi = p.x + 0.5f * sw;
    float ylo = p.y - 0.5f * sh, yhi = p.y + 0.5f * sh;
    int bx0 = (int)floorf(xlo);
    int by0 = (int)floorf(ylo);

    float ox[KCOV], oy[KCOV];
    int   bx[KCOV], by[KCOV];
#pragma unroll
    for (int k = 0; k < KCOV; ++k) {
        bx[k] = bx0 + k;
        by[k] = by0 + k;
        float blx = (float)bx[k];
        float bly = (float)by[k];
        float ovx = fminf(blx + 1.0f, xhi) - fmaxf(blx, xlo);
        float ovy = fminf(bly + 1.0f, yhi) - fmaxf(bly, ylo);
        // out-of-range bins contribute exactly zero (same result as the
        // reference's clip+where; we additionally never touch memory for them)
        ox[k] = ((unsigned)bx[k] < (unsigned)NBX) ? fmaxf(ovx, 0.0f) : 0.0f;
        oy[k] = ((unsigned)by[k] < (unsigned)NBY) ? fmaxf(ovy, 0.0f) : 0.0f;
    }

#pragma unroll
    for (int k = 0; k < KCOV; ++k) {
        if (ox[k] > 0.0f) {
            float cx = dens * ox[k];
            int base = bx[k] * NBY;
#pragma unroll
            for (int l = 0; l < KCOV; ++l) {
                if (oy[l] > 0.0f) {
                    // hardware f32 L2 atomic, no-return form (TH non-returning)
                    unsafeAtomicAdd(&grid[base + by[l]], cx * oy[l]);
                }
            }
        }
    }
}

// Vectorized zeroing: one b128 store per thread, no tail logic in-kernel.
__global__ __launch_bounds__(256) void PinUtil_zero4_kernel(float4* __restrict__ g, int n4) {
    int i = blockIdx.x * blockDim.x + threadIdx.x;
    if (i < n4) g[i] = make_float4(0.f, 0.f, 0.f, 0.f);
}

// Scalar tail (generic safety; never launched for 512*512).
__global__ __launch_bounds__(256) void PinUtil_zero1_kernel(float* __restrict__ g, int lo, int n) {
    int i = lo + blockIdx.x * blockDim.x + threadIdx.x;
    if (i < n) g[i] = 0.f;
}

extern "C" void kernel_launch(void* const* d_in, const int* in_sizes, int n_in,
                              void* d_out, int out_size, void* d_ws, size_t ws_size,
                              hipStream_t stream) {
    (void)n_in; (void)d_ws; (void)ws_size;
    const float2* sizes = (const float2*)d_in[0];  // [N,2] f32
    const float2* pos   = (const float2*)d_in[1];  // [N,2] f32
    const float*  pinw  = (const float*)d_in[2];   // [N]   f32
    float* grid = (float*)d_out;                   // [512*512] f32
    int n = in_sizes[2];                           // N instances

    int n4 = out_size >> 2;
    if (n4 > 0) {
        PinUtil_zero4_kernel<<<(n4 + 255) / 256, 256, 0, stream>>>((float4*)grid, n4);
    }
    int tail = out_size - (n4 << 2);
    if (tail > 0) {
        PinUtil_zero1_kernel<<<(tail + 255) / 256, 256, 0, stream>>>(grid, n4 << 2, out_size);
    }

    int blocks = (n + 255) / 256;
    PinUtil_scatter_kernel<<<blocks, 256, 0, stream>>>(sizes, pos, pinw, grid, n);
}